// ROIPooling_41283225649254
// MI455X (gfx1250) — compile-verified
//
#include <hip/hip_runtime.h>
#include <math.h>
#include <stdint.h>

#define OUTP 7
#define CH   256
#define HH   50
#define WW   50

// input staging: double-buffered, 2 x 7500 floats = 60 KB
#define BUF_ELEMS 7500
// output staging: double-buffered, 2 x (8*49) floats ~ 3.1 KB
#define OBUF_ELEMS (8 * 49)

typedef unsigned int v4u __attribute__((ext_vector_type(4)));
typedef int          v8i __attribute__((ext_vector_type(8)));
typedef int          v4i __attribute__((ext_vector_type(4)));

__device__ __forceinline__ int imin(int a, int b) { return a < b ? a : b; }
__device__ __forceinline__ int imax(int a, int b) { return a > b ? a : b; }

// Pack D# groups for a (up to) 3-D tile [d2, d1, d0] of 4-byte elements.
// stride0/stride1 in elements. Returns via out-params.
__device__ __forceinline__ void tdm_make_desc(uint32_t lds_byte_addr,
                                              uint64_t gaddr,
                                              int d0, int d1, int d2,
                                              uint64_t stride0, uint64_t stride1,
                                              v4u& g0, v8i& g1, v4i& g2, v4i& g3)
{
    // ---- Group 0 (128b): count=1 | lds_addr | global_addr[56:0] | type=2
    g0.x = 1u;                                            // count=1, user desc
    g0.y = lds_byte_addr;                                 // lds_addr (bytes)
    g0.z = (uint32_t)(gaddr & 0xFFFFFFFFull);             // global_addr[31:0]
    g0.w = (uint32_t)((gaddr >> 32) & 0x01FFFFFFull)      // global_addr[56:32]
         | (2u << 30);                                    // type = 2 ("image")

    // ---- Group 1 (256b); tensor dims == tile dims so no OOB path triggers
    uint32_t td0 = (uint32_t)d0;
    uint32_t td1 = (uint32_t)imax(d1, 1);
    g1[0] = (int)(2u << 16);              // workgroup_mask=0, data_size=2 (4B)
    g1[1] = (int)((td0 & 0xFFFFu) << 16); // atomic_barrier_addr=0 | td0 lo16
    g1[2] = (int)((td0 >> 16) | ((td1 & 0xFFFFu) << 16));
    g1[3] = (int)((td1 >> 16) | (((uint32_t)d0 & 0xFFFFu) << 16));  // tile_dim0
    g1[4] = (int)(((uint32_t)d1 & 0xFFFFu) |
                  (((uint32_t)d2 & 0xFFFFu) << 16));      // tile_dim1, tile_dim2
    g1[5] = (int)(uint32_t)(stride0 & 0xFFFFFFFFull);     // dim0_stride lo32
    g1[6] = (int)((uint32_t)(stride0 >> 32) |
                  ((uint32_t)(stride1 & 0xFFFFull) << 16)); // d0s hi16|d1s lo16
    g1[7] = (int)(uint32_t)(stride1 >> 16);               // d1s bits [47:16]

    // ---- Group 2: tensor_dim2 (>= tile_dim2), rest unused (iterate off)
    g2[0] = imax(d2, 1); g2[1] = 0; g2[2] = 0; g2[3] = 0;
    // ---- Group 3: unused
    g3[0] = 0; g3[1] = 0; g3[2] = 0; g3[3] = 0;
}

__device__ __forceinline__ void tdm_load_3d(uint32_t lds_byte_addr, uint64_t gaddr,
                                            int wx, int wy, int cc,
                                            uint64_t stride0, uint64_t stride1)
{
    v4u g0; v8i g1; v4i g2; v4i g3;
    tdm_make_desc(lds_byte_addr, gaddr, wx, wy, cc, stride0, stride1,
                  g0, g1, g2, g3);
#if defined(__clang_major__) && (__clang_major__ >= 23)
    v8i g4; g4[0]=0; g4[1]=0; g4[2]=0; g4[3]=0; g4[4]=0; g4[5]=0; g4[6]=0; g4[7]=0;
    __builtin_amdgcn_tensor_load_to_lds(g0, g1, g2, g3, g4, 0);
#else
    __builtin_amdgcn_tensor_load_to_lds(g0, g1, g2, g3, 0);
#endif
}

__device__ __forceinline__ void tdm_store_1d(uint32_t lds_byte_addr, uint64_t gaddr,
                                             int nelems)
{
    v4u g0; v8i g1; v4i g2; v4i g3;
    // 1-D contiguous tile: tile_dim0 = nelems (<= 392, fits 16 bits)
    tdm_make_desc(lds_byte_addr, gaddr, nelems, 1, 1,
                  (uint64_t)nelems, (uint64_t)nelems, g0, g1, g2, g3);
#if defined(__clang_major__) && (__clang_major__ >= 23)
    v8i g4; g4[0]=0; g4[1]=0; g4[2]=0; g4[3]=0; g4[4]=0; g4[5]=0; g4[6]=0; g4[7]=0;
    __builtin_amdgcn_tensor_store_from_lds(g0, g1, g2, g3, g4, 0);
#else
    __builtin_amdgcn_tensor_store_from_lds(g0, g1, g2, g3, 0);
#endif
}

__global__ __launch_bounds__(256)
void ROIPooling_41283225649254_kernel(const float* __restrict__ feat,
                                      const float* __restrict__ boxes,
                                      const int*   __restrict__ p_imgsize,
                                      float*       __restrict__ out,
                                      int N)
{
    __shared__ float smem[2][BUF_ELEMS];
    __shared__ float obuf[2][OBUF_ELEMS];

    const int roi = blockIdx.x;
    if (roi >= N) return;

    const float image_size = (float)p_imgsize[0];
    const float scale_h = (float)HH / image_size;
    const float scale_w = (float)WW / image_size;

    // ---- ROI integer coords: exact replica of reference clamping logic
    const float bx = boxes[roi * 4 + 0];
    const float by = boxes[roi * 4 + 1];
    const float bw = boxes[roi * 4 + 2];
    const float bh = boxes[roi * 4 + 3];
    const bool bad = (bw <= 0.0f) || (bh <= 0.0f);
    const float x1f = (bad ? 0.25f * image_size : bx)      * scale_w;
    const float x2f = (bad ? 0.75f * image_size : bx + bw) * scale_w;
    const float y1f = (bad ? 0.25f * image_size : by)      * scale_h;
    const float y2f = (bad ? 0.75f * image_size : by + bh) * scale_h;

    int x1 = imax(0, (int)x1f);            // trunc-toward-zero == astype(int32)
    int y1 = imax(0, (int)y1f);
    int x2 = imin(WW, (int)x2f + 1);
    int y2 = imin(HH, (int)y2f + 1);
    if (x2 <= x1 + 1) x2 = imin(x1 + 2, WW);
    if (y2 <= y1 + 1) y2 = imin(y1 + 2, HH);
    x1 = imin(imax(x1, 0), WW - 2);
    y1 = imin(imax(y1, 0), HH - 2);
    x2 = imax(x1 + 1, imin(x2, WW));
    y2 = imax(y1 + 1, imin(y2, HH));

    const int wx  = x2 - x1;
    const int wy  = y2 - y1;
    const int per = wx * wy;               // elems per channel plane (<= 2500)
    int CC = BUF_ELEMS / per;              // channels per chunk (>= 3)
    if (CC > 8) CC = 8;
    const int nchunks = (CH + CC - 1) / CC;

    const int tid = threadIdx.x;
    const uint64_t gbase = (uint64_t)(uintptr_t)feat;
    const uint64_t obase = (uint64_t)(uintptr_t)(out + (size_t)roi * CH * 49);

    // Prime: wave 0 issues the DMA for chunk 0
    if (tid < 32) {
        const int cc0 = imin(CC, CH);
        const uint64_t g = gbase + 4ull * ((uint64_t)y1 * WW + (uint64_t)x1);
        tdm_load_3d((uint32_t)(uintptr_t)&smem[0][0], g, wx, wy, cc0,
                    (uint64_t)WW, (uint64_t)(HH * WW));
    }

    // In-order TDM queue from wave 0: load0, load1, store0, load2, store1, ...
    // At iter i, after issuing load(i+1), ops younger than load(i) are at most
    // {store(i-1), load(i+1)} -> wait_tensorcnt <= (i>=1)+(i+1<nchunks)
    // guarantees load(i) done AND store(i-2) done (staging buffer reusable).
    for (int i = 0; i < nchunks; ++i) {
        const int cb     = i * CC;
        const int ccount = imin(CC, CH - cb);

        if (tid < 32) {
            if (i + 1 < nchunks) {
                const int cb2 = (i + 1) * CC;
                const int cc2 = imin(CC, CH - cb2);
                const uint64_t g = gbase +
                    4ull * ((uint64_t)cb2 * (HH * WW) +
                            (uint64_t)y1 * WW + (uint64_t)x1);
                tdm_load_3d((uint32_t)(uintptr_t)&smem[(i + 1) & 1][0], g,
                            wx, wy, cc2, (uint64_t)WW, (uint64_t)(HH * WW));
                if (i >= 1) __builtin_amdgcn_s_wait_tensorcnt(2);
                else        __builtin_amdgcn_s_wait_tensorcnt(1);
            } else {
                if (i >= 1) __builtin_amdgcn_s_wait_tensorcnt(1);
                else        __builtin_amdgcn_s_wait_tensorcnt(0);
            }
        }
        __syncthreads();   // chunk i feature data visible to all waves

        const float* buf = &smem[i & 1][0];
        float*       ob  = &obuf[i & 1][0];
        const int items = ccount * (OUTP * OUTP);
        for (int t = tid; t < items; t += 256) {
            const int cl  = t / 49;
            const int bin = t - cl * 49;
            const int oy  = bin / 7;
            const int ox  = bin - oy * 7;
            // bin [s,e): s = floor(k*sz/7), e = ceil((k+1)*sz/7); never empty
            const int sx = (ox * wx) / OUTP;
            const int ex = ((ox + 1) * wx + OUTP - 1) / OUTP;
            const int sy = (oy * wy) / OUTP;
            const int ey = ((oy + 1) * wy + OUTP - 1) / OUTP;

            const float* p = buf + cl * per;
            float m = -INFINITY;
            for (int yy = sy; yy < ey; ++yy) {
                const float* row = p + yy * wx;
                for (int xx = sx; xx < ex; ++xx) m = fmaxf(m, row[xx]);
            }
            ob[t] = m;     // staged: linear order matches global layout
        }
        __syncthreads();   // all pooled results for chunk i are in LDS

        // Wave 0 DMAs the contiguous [ccount*49] result run to global memory
        if (tid < 32) {
            tdm_store_1d((uint32_t)(uintptr_t)ob,
                         obase + 4ull * (uint64_t)(cb * 49),
                         ccount * 49);
        }
        // NOTE: no extra barrier needed here; next iteration's first
        // __syncthreads orders compute reads, and the tensorcnt wait protects
        // both LDS staging buffers before reuse. S_ENDPGM's implicit wait-idle
        // drains the final store.
    }
}

extern "C" void kernel_launch(void* const* d_in, const int* in_sizes, int n_in,
                              void* d_out, int out_size, void* d_ws, size_t ws_size,
                              hipStream_t stream) {
    const float* feat  = (const float*)d_in[0];   // [1,256,50,50] f32
    const float* boxes = (const float*)d_in[1];   // [1,256,4]     f32
    const int*   imgsz = (const int*)d_in[2];     // scalar int (800)
    float* out = (float*)d_out;                   // [256,256,7,7] f32

    const int N = in_sizes[1] / 4;                // number of ROIs (256)
    ROIPooling_41283225649254_kernel<<<N, 256, 0, stream>>>(feat, boxes, imgsz,
                                                            out, N);
}